// RPNBoxSelector_9062380995328
// MI455X (gfx1250) — compile-verified
//
#include <hip/hip_runtime.h>
#include <hip/hip_bf16.h>

// ---------------- problem constants (match reference) ----------------
#define N_IMG 8
#define A_    3
#define H_    100
#define W_    100
#define NBOX  (H_ * W_ * A_)   // 30000
#define PRE   2000
#define POST  1000
#define NMS_THRESH 0.7f
#define BBOX_CLIP  4.135166556742356f   // log(1000/16)
#define SORT_N 32768
#define NWORDS 63                       // ceil(2000/32)

typedef unsigned int  v4u __attribute__((ext_vector_type(4)));
typedef int           v8i __attribute__((ext_vector_type(8)));
typedef int           v4i __attribute__((ext_vector_type(4)));

// =====================================================================
// Kernel 1: per-image top-2000 via full in-LDS bitonic sort (256KB LDS,
// only possible on CDNA5's 320KB WGP LDS). Keys: sortable score bits in
// high 32, (0xFFFFFFFF - index) low 32 -> descending sort gives exact
// jax.lax.top_k order (score desc, index asc tiebreak).
// =====================================================================
__global__ __launch_bounds__(1024) void topk_kernel(const float* __restrict__ obj,
                                                    int* __restrict__ topIdx,
                                                    float* __restrict__ topScore) {
    __shared__ unsigned long long keys[SORT_N];   // 256 KB
    const int n   = blockIdx.x;
    const int tid = threadIdx.x;
    const float* o = obj + (size_t)n * (A_ * H_ * W_);

    for (int i = tid; i < SORT_N; i += 1024) {
        unsigned long long key = 0ull;
        if (i < NBOX) {
            // scores index = (h*W + w)*A + a   (NHWA layout)
            int a = i % A_;
            int w = (i / A_) % W_;
            int h = i / (A_ * W_);
            float x = o[((size_t)a * H_ + h) * W_ + w];
            unsigned ub = __float_as_uint(x);
            unsigned s  = (ub & 0x80000000u) ? ~ub : (ub | 0x80000000u);
            key = ((unsigned long long)s << 32) | (0xFFFFFFFFu - (unsigned)i);
        }
        keys[i] = key;
    }
    __syncthreads();

    // bitonic sort, descending
    for (unsigned k2 = 2; k2 <= SORT_N; k2 <<= 1) {
        for (unsigned j = k2 >> 1; j > 0; j >>= 1) {
            for (unsigned i = tid; i < SORT_N; i += 1024) {
                unsigned ixj = i ^ j;
                if (ixj > i) {
                    unsigned long long a = keys[i];
                    unsigned long long b = keys[ixj];
                    bool up = ((i & k2) == 0);
                    if (up ? (a < b) : (a > b)) { keys[i] = b; keys[ixj] = a; }
                }
            }
            __syncthreads();
        }
    }

    for (int r = tid; r < PRE; r += 1024) {
        unsigned long long key = keys[r];
        unsigned idx = 0xFFFFFFFFu - (unsigned)(key & 0xFFFFFFFFull);
        unsigned s   = (unsigned)(key >> 32);
        unsigned ub  = (s & 0x80000000u) ? (s ^ 0x80000000u) : ~s;
        float x = __uint_as_float(ub);
        topIdx[n * PRE + r]   = (int)idx;
        topScore[n * PRE + r] = 1.0f / (1.0f + expf(-x));   // sigmoid (monotonic; sorted on raw)
    }
}

// =====================================================================
// Kernel 2: gather deltas/anchors at top indices, decode, clip, validity
// =====================================================================
__global__ void decode_kernel(const float* __restrict__ br,
                              const float* __restrict__ anc,
                              const int* __restrict__ ihp,
                              const int* __restrict__ iwp,
                              const int* __restrict__ topIdx,
                              float* __restrict__ props,
                              int* __restrict__ valid) {
    int g = blockIdx.x * blockDim.x + threadIdx.x;
    if (g >= N_IMG * PRE) return;
    int n = g / PRE;
    float imw = (float)iwp[0];
    float imh = (float)ihp[0];

    int idx = topIdx[g];
    int a = idx % A_;
    int w = (idx / A_) % W_;
    int h = idx / (A_ * W_);

    const float* brn = br + (size_t)n * (A_ * 4 * H_ * W_);
    float d0 = brn[(((size_t)a * 4 + 0) * H_ + h) * W_ + w];
    float d1 = brn[(((size_t)a * 4 + 1) * H_ + h) * W_ + w];
    float d2 = brn[(((size_t)a * 4 + 2) * H_ + h) * W_ + w];
    float d3 = brn[(((size_t)a * 4 + 3) * H_ + h) * W_ + w];

    const float* an = anc + ((size_t)n * NBOX + idx) * 4;
    float x1 = an[0], y1 = an[1], x2 = an[2], y2 = an[3];
    float wd = x2 - x1 + 1.0f;
    float ht = y2 - y1 + 1.0f;
    float cx = x1 + 0.5f * wd;
    float cy = y1 + 0.5f * ht;
    float dw = fminf(d2, BBOX_CLIP);
    float dh = fminf(d3, BBOX_CLIP);
    float pcx = d0 * wd + cx;
    float pcy = d1 * ht + cy;
    float pw  = expf(dw) * wd;
    float ph  = expf(dh) * ht;
    float px1 = pcx - 0.5f * pw;
    float py1 = pcy - 0.5f * ph;
    float px2 = pcx + 0.5f * pw - 1.0f;
    float py2 = pcy + 0.5f * ph - 1.0f;
    px1 = fminf(fmaxf(px1, 0.0f), imw - 1.0f);
    px2 = fminf(fmaxf(px2, 0.0f), imw - 1.0f);
    py1 = fminf(fmaxf(py1, 0.0f), imh - 1.0f);
    py2 = fminf(fmaxf(py2, 0.0f), imh - 1.0f);

    float ws2 = px2 - px1 + 1.0f;
    float hs  = py2 - py1 + 1.0f;
    float xc  = px1 + 0.5f * ws2;
    float yc  = py1 + 0.5f * hs;
    int v = (ws2 >= 0.0f) && (hs >= 0.0f) && (xc < imw) && (yc < imh);  // MIN_SIZE == 0

    props[(size_t)g * 4 + 0] = px1;
    props[(size_t)g * 4 + 1] = py1;
    props[(size_t)g * 4 + 2] = px2;
    props[(size_t)g * 4 + 3] = py2;
    valid[g] = v;
}

// =====================================================================
// Kernel 3: suppression bitmask. One block per image; proposals (32KB)
// staged into LDS with a single Tensor-Data-Mover DMA (CDNA5
// tensor_load_to_lds, TENSORcnt tracked, s_wait_tensorcnt fence).
// mask[n][i][w] bit b set <=> j=w*32+b, j>i, iou(i,j)>0.7
// =====================================================================
__global__ __launch_bounds__(1024) void iou_mask_kernel(const float* __restrict__ props,
                                                        unsigned* __restrict__ mask) {
    __shared__ float shBox[PRE * 4];   // 32 KB
    const int n   = blockIdx.x;
    const int tid = threadIdx.x;
    const float* p = props + (size_t)n * PRE * 4;

    if (tid == 0) {
        // --- Tensor DMA Descriptor (ISA cdna5 sec 8.3/8.4) ---
        unsigned long long gaddr = (unsigned long long)(const void*)p;
        unsigned ldsOff = (unsigned)(unsigned long long)(void*)&shBox[0]; // flat LDS addr: low 32 bits = LDS byte offset
        const unsigned nElem = PRE * 4;   // 8000 x 4B elements

        v4u g0;
        g0.x = 1u;                                                     // count=1 (valid descriptor)
        g0.y = ldsOff;                                                 // lds_addr
        g0.z = (unsigned)gaddr;                                        // global_addr[31:0]
        g0.w = (unsigned)((gaddr >> 32) & 0x01FFFFFFull) | (2u << 30); // global_addr[56:32] | type=2
        v8i g1;
        g1[0] = (int)(2u << 16);                 // data_size=2 (4 bytes), no pad/iterate/multicast
        g1[1] = (int)((nElem & 0xFFFFu) << 16);  // tensor_dim0[15:0]
        g1[2] = (int)(((nElem >> 16) & 0xFFFFu) | (1u << 16)); // tensor_dim0[31:16], tensor_dim1=1
        g1[3] = (int)((nElem & 0xFFFFu) << 16);  // tensor_dim1[31:16]=0, tile_dim0=8000
        g1[4] = 0;                               // tile_dim1=0 (1D), tile_dim2=0
        g1[5] = (int)nElem;                      // tensor_dim0_stride[31:0]
        g1[6] = 0;                               // stride0[47:32]=0, stride1[15:0]=0
        g1[7] = 0;
        v4i g2 = {0, 0, 0, 0};                   // groups 2/3 unused (<=2D tensor)
        v4i g3 = {0, 0, 0, 0};
        v8i g4 = {0, 0, 0, 0, 0, 0, 0, 0};       // extra group (clang-23 6-arg form), zero-filled
        __builtin_amdgcn_tensor_load_to_lds(g0, g1, g2, g3, g4, 0);
    }
    __builtin_amdgcn_s_wait_tensorcnt(0);
    __syncthreads();

    unsigned* mrow = mask + (size_t)n * PRE * NWORDS;
    for (int i = tid; i < PRE; i += 1024) {
        float ix1 = shBox[i * 4 + 0];
        float iy1 = shBox[i * 4 + 1];
        float ix2 = shBox[i * 4 + 2];
        float iy2 = shBox[i * 4 + 3];
        float areaI = (ix2 - ix1 + 1.0f) * (iy2 - iy1 + 1.0f);
        for (int wd = 0; wd < NWORDS; ++wd) {
            unsigned bits = 0u;
            int jbase = wd * 32;
            for (int b = 0; b < 32; ++b) {
                int j = jbase + b;
                if (j < PRE && j > i) {
                    float jx1 = shBox[j * 4 + 0];
                    float jy1 = shBox[j * 4 + 1];
                    float jx2 = shBox[j * 4 + 2];
                    float jy2 = shBox[j * 4 + 3];
                    float xx1 = fmaxf(ix1, jx1);
                    float yy1 = fmaxf(iy1, jy1);
                    float xx2 = fminf(ix2, jx2);
                    float yy2 = fminf(iy2, jy2);
                    float iw = fmaxf(xx2 - xx1 + 1.0f, 0.0f);
                    float ih = fmaxf(yy2 - yy1 + 1.0f, 0.0f);
                    float inter = iw * ih;
                    float areaJ = (jx2 - jx1 + 1.0f) * (jy2 - jy1 + 1.0f);
                    float iou = inter / (areaI + areaJ - inter);
                    if (iou > NMS_THRESH) bits |= (1u << b);
                }
            }
            mrow[(size_t)i * NWORDS + wd] = bits;
        }
    }
}

// =====================================================================
// Kernel 4: sequential greedy-NMS scan on the bitmask (exact semantics of
// the reference fori_loop), then rank-compaction into d_out.
// =====================================================================
__global__ __launch_bounds__(128) void nms_scan_kernel(const float* __restrict__ props,
                                                       const float* __restrict__ topScore,
                                                       const int* __restrict__ valid,
                                                       const unsigned* __restrict__ mask,
                                                       float* __restrict__ outBoxes,
                                                       float* __restrict__ outScores,
                                                       float* __restrict__ outValid) {
    __shared__ unsigned keep[NWORDS];
    __shared__ int off[NWORDS + 1];
    const int n   = blockIdx.x;
    const int tid = threadIdx.x;

    for (int w = tid; w < NWORDS; w += blockDim.x) {
        unsigned m = 0u;
        for (int b = 0; b < 32; ++b) {
            int j = w * 32 + b;
            if (j < PRE && valid[n * PRE + j]) m |= (1u << b);
        }
        keep[w] = m;
    }
    __syncthreads();

    for (int i = 0; i < PRE; ++i) {
        bool alive = (keep[i >> 5] >> (i & 31)) & 1u;   // uniform across block
        __syncthreads();
        if (alive) {
            const unsigned* row = mask + ((size_t)n * PRE + i) * NWORDS;
            for (int w = tid; w < NWORDS; w += blockDim.x) keep[w] &= ~row[w];
        }
        __syncthreads();
    }

    if (tid == 0) {
        int acc = 0;
        for (int w = 0; w < NWORDS; ++w) { off[w] = acc; acc += __popc(keep[w]); }
        off[NWORDS] = acc;
    }
    __syncthreads();
    int total = off[NWORDS];
    int nvalid = total < POST ? total : POST;

    // zero-fill this image's output region (d_out is poisoned by harness)
    for (int k = tid; k < POST * 4; k += blockDim.x) outBoxes[(size_t)n * POST * 4 + k] = 0.0f;
    for (int k = tid; k < POST; k += blockDim.x) {
        outScores[n * POST + k] = 0.0f;
        outValid[n * POST + k]  = (k < nvalid) ? 1.0f : 0.0f;
    }
    __syncthreads();

    for (int w = tid; w < NWORDS; w += blockDim.x) {
        unsigned m = keep[w];
        int rank = off[w];
        while (m) {
            int b = __ffs(m) - 1;
            m &= m - 1;
            int i = w * 32 + b;
            if (rank < POST) {
                const float* src = props + ((size_t)n * PRE + i) * 4;
                float* dst = outBoxes + ((size_t)n * POST + rank) * 4;
                dst[0] = src[0]; dst[1] = src[1]; dst[2] = src[2]; dst[3] = src[3];
                outScores[n * POST + rank] = topScore[n * PRE + i];
            }
            ++rank;
        }
    }
}

// =====================================================================
extern "C" void kernel_launch(void* const* d_in, const int* in_sizes, int n_in,
                              void* d_out, int out_size, void* d_ws, size_t ws_size,
                              hipStream_t stream) {
    const float* objectness = (const float*)d_in[0];   // (8,3,100,100)
    const float* box_reg    = (const float*)d_in[1];   // (8,12,100,100)
    const float* anchors    = (const float*)d_in[2];   // (8,30000,4)
    const int*   img_h      = (const int*)d_in[3];
    const int*   img_w      = (const int*)d_in[4];

    char* ws = (char*)d_ws;
    int*      topIdx   = (int*)(ws);                         //  64,000 B
    float*    topScore = (float*)(ws + 64000);               //  64,000 B
    float*    props    = (float*)(ws + 128000);              // 256,000 B
    int*      valid    = (int*)(ws + 384000);                //  64,000 B
    unsigned* mask     = (unsigned*)(ws + 448000);           // 4,032,000 B

    float* out       = (float*)d_out;
    float* outBoxes  = out;                    // 8*1000*4
    float* outScores = out + N_IMG * POST * 4; // 8*1000
    float* outValid  = out + N_IMG * POST * 5; // 8*1000

    topk_kernel<<<N_IMG, 1024, 0, stream>>>(objectness, topIdx, topScore);

    int tot = N_IMG * PRE;
    decode_kernel<<<(tot + 255) / 256, 256, 0, stream>>>(box_reg, anchors, img_h, img_w,
                                                         topIdx, props, valid);

    iou_mask_kernel<<<N_IMG, 1024, 0, stream>>>(props, mask);

    nms_scan_kernel<<<N_IMG, 128, 0, stream>>>(props, topScore, valid, mask,
                                               outBoxes, outScores, outValid);
}